// MDense_21242908246016
// MI455X (gfx1250) — compile-verified
//
#include <hip/hip_runtime.h>
#include <hip/hip_bf16.h>

typedef __attribute__((ext_vector_type(2))) float v2f;
typedef __attribute__((ext_vector_type(8))) float v8f;

#define UNITS 256
#define IN_DIM 16
#define CH 2
#define R0 64
#define R1 8
#define R2 2
#define WCOLS (UNITS * CH)   // 512

__device__ __forceinline__ float fast_tanh(float x) {
#if __has_builtin(__builtin_amdgcn_tanhf)
    return __builtin_amdgcn_tanhf(x);
#elif __has_builtin(__builtin_amdgcn_tanh_f32)
    return __builtin_amdgcn_tanh_f32(x);
#else
    x = fminf(fmaxf(x, -15.0f), 15.0f);
    float e2 = __expf(2.0f * x);
    return (e2 - 1.0f) / (e2 + 1.0f);
#endif
}

// ---------------------------------------------------------------------------
// Stage 1: Tucker reconstruction of the weight matrix.
// Wmat[k][n], k = in_dim index (0..15), n = c*256 + u (0..511), row stride 512.
// One thread per (u, i, c) output element; factored contraction:
//   W[u,i,c] = sum_r0 f0[u,r0] * ( sum_r1 f1[i,r1] * ( sum_r2 core[r0,r1,r2]*f2[c,r2] ) )
// core reads are wave-uniform -> scalar loads; ~1.5K FMA per thread, negligible.
// ---------------------------------------------------------------------------
__global__ __launch_bounds__(512)
void mdense_build_w(const float* __restrict__ core,
                    const float* __restrict__ f0,
                    const float* __restrict__ f1,
                    const float* __restrict__ f2,
                    float* __restrict__ Wmat) {
    int t = blockIdx.x * 512 + threadIdx.x;      // 0 .. 8191
    int c = t & 1;
    int i = (t >> 1) & 15;
    int u = t >> 5;

    float s2a = f2[c * R2 + 0];
    float s2b = f2[c * R2 + 1];

    float acc = 0.0f;
    for (int r0 = 0; r0 < R0; ++r0) {
        float g = 0.0f;
        #pragma unroll
        for (int r1 = 0; r1 < R1; ++r1) {
            const float* cp = core + ((r0 * R1 + r1) * R2);
            float h = fmaf(cp[0], s2a, cp[1] * s2b);
            g = fmaf(f1[i * R1 + r1], h, g);
        }
        acc = fmaf(f0[u * R0 + r0], g, acc);
    }
    Wmat[i * WCOLS + c * UNITS + u] = acc;
}

// ---------------------------------------------------------------------------
// Stage 2: main fused GEMM + tanh epilogue.
// One wave -> one 16x16 (batch x units) output tile, via two f32 WMMA
// accumulator tiles (channel 0 at columns n = u0..u0+15, channel 1 at n+256).
// K = 16 split into 4 k-steps of 4 -> 8 x v_wmma_f32_16x16x4_f32 per wave.
// ---------------------------------------------------------------------------
__global__ __launch_bounds__(128)
void mdense_main(const float* __restrict__ X,      // (nbatch, 16)
                 const float* __restrict__ Wmat,   // (16, 512)
                 const float* __restrict__ bias,   // (256, 2)
                 const float* __restrict__ factor, // (256, 2)
                 float* __restrict__ out,          // (nbatch, 256)
                 int nbatch) {
    const int lane = threadIdx.x & 31;
    const int wave = threadIdx.x >> 5;
    const int tile  = blockIdx.x * 4 + wave;
    const int utile = tile & 15;        // 16 unit tiles
    const int btile = tile >> 4;        // batch tiles
    const int b0 = btile * 16;
    const int u0 = utile * 16;
    if (b0 >= nbatch) return;           // wave-uniform: EXEC stays all-ones

    const int half = lane >> 4;         // 0: lanes 0-15, 1: lanes 16-31
    const int l16  = lane & 15;

    // ---- A fragments: lane holds A[row = l16][k = 4*kk + 2*half + {0,1}] ----
    const float* arow = X + (size_t)(b0 + l16) * IN_DIM + 2 * half;
    v2f afrag[4];
    #pragma unroll
    for (int kk = 0; kk < 4; ++kk) {
        afrag[kk] = *(const v2f*)(arow + 4 * kk);
    }

    // ---- B fragments: VGPR v, lane holds B[k = 4*kk + 2*half + v][n0 + l16] ----
    v2f bfrag[2][4];
    #pragma unroll
    for (int c = 0; c < 2; ++c) {
        const int ncol = c * UNITS + u0 + l16;
        #pragma unroll
        for (int kk = 0; kk < 4; ++kk) {
            const int kb = 4 * kk + 2 * half;
            v2f b;
            b.x = Wmat[(kb + 0) * WCOLS + ncol];
            b.y = Wmat[(kb + 1) * WCOLS + ncol];
            bfrag[c][kk] = b;
        }
    }

    // ---- 8x WMMA: D = A x B + C, f32 exact ----
    v8f acc0 = {};
    v8f acc1 = {};
    #pragma unroll
    for (int kk = 0; kk < 4; ++kk) {
        acc0 = __builtin_amdgcn_wmma_f32_16x16x4_f32(
            false, afrag[kk], false, bfrag[0][kk], (short)0, acc0, false, false);
        acc1 = __builtin_amdgcn_wmma_f32_16x16x4_f32(
            false, afrag[kk], false, bfrag[1][kk], (short)0, acc1, false, false);
    }

    // ---- fused epilogue: tanh(acc + bias) * factor, summed over channels ----
    const int u = u0 + l16;
    const float bs0 = bias[u * CH + 0];
    const float bs1 = bias[u * CH + 1];
    const float fs0 = factor[u * CH + 0];
    const float fs1 = factor[u * CH + 1];

    // C/D layout: VGPR r, this lane -> M = r + 8*half, N = l16
    float* orow = out + (size_t)(b0 + 8 * half) * UNITS + u;
    #pragma unroll
    for (int r = 0; r < 8; ++r) {
        float v = fast_tanh(acc0[r] + bs0) * fs0 + fast_tanh(acc1[r] + bs1) * fs1;
        orow[(size_t)r * UNITS] = v;
    }
}

extern "C" void kernel_launch(void* const* d_in, const int* in_sizes, int n_in,
                              void* d_out, int out_size, void* d_ws, size_t ws_size,
                              hipStream_t stream) {
    const float* X      = (const float*)d_in[0];   // (N, 16)
    const float* core   = (const float*)d_in[1];   // (64, 8, 2)
    const float* f0     = (const float*)d_in[2];   // (256, 64)
    const float* f1     = (const float*)d_in[3];   // (16, 8)
    const float* f2     = (const float*)d_in[4];   // (2, 2)
    const float* bias   = (const float*)d_in[5];   // (256, 2)
    const float* factor = (const float*)d_in[6];   // (256, 2)
    float* out = (float*)d_out;

    float* Wmat = (float*)d_ws;                    // 16*512*4 = 32 KB scratch

    const int nbatch = in_sizes[0] / IN_DIM;       // 262144

    // Stage 1: build W (8192 threads)
    mdense_build_w<<<16, 512, 0, stream>>>(core, f0, f1, f2, Wmat);

    // Stage 2: tiles = (nbatch/16) * 16 unit-tiles; 4 waves (tiles) per block
    const int total_tiles = (nbatch / 16) * 16;
    const int blocks = (total_tiles + 3) / 4;
    mdense_main<<<blocks, 128, 0, stream>>>(X, Wmat, bias, factor, out, nbatch);
}